// Codebook_69698729280154
// MI455X (gfx1250) — compile-verified
//
#include <hip/hip_runtime.h>

typedef float v2f __attribute__((ext_vector_type(2)));
typedef float v8f __attribute__((ext_vector_type(8)));

#define DDIM    256
#define KCODES  1024
#define HWSZ    4096          // 64*64
#define NROWS   65536         // 16*4096
#define NELEM   16777216      // 16*256*64*64
#define CHUNK   32            // codes staged in LDS at a time
#define LSTRIDE 258           // 256 + 2 pad floats -> bank-conflict-free
#define BETA_F  0.25f

// ---------------- Phase 0: half-norms of codebook rows + zero loss slot ----
__global__ __launch_bounds__(256) void vq_norms_kernel(
    const float* __restrict__ emb, float* __restrict__ norms,
    float* __restrict__ loss_out)
{
    int c = blockIdx.x * 256 + threadIdx.x;
    if (c == 0) loss_out[0] = 0.0f;
    if (c < KCODES) {
        const float4* e = (const float4*)(emb + (size_t)c * DDIM);
        float s = 0.0f;
        #pragma unroll 8
        for (int i = 0; i < DDIM / 4; ++i) {
            float4 v = e[i];
            s += v.x * v.x + v.y * v.y + v.z * v.z + v.w * v.w;
        }
        norms[c] = 0.5f * s;
    }
}

// ------- Phase 1: fused GEMM (WMMA f32 16x16x4) + argmin + loss ------------
// Per row: best score v = 0.5||e||^2 - z.e  =>  ||zq - z||^2 = ||z||^2 + 2v.
// So the commitment loss is fully determined here; phase 2 never reads z.
__global__ __launch_bounds__(256) void vq_argmin_kernel(
    const float* __restrict__ z, const float* __restrict__ emb,
    const float* __restrict__ norms, int* __restrict__ idx_ws,
    float* __restrict__ idx_out, float* __restrict__ loss_out)
{
    __shared__ float Bsh[CHUNK * LSTRIDE];   // 32 codes * 258 floats = 33KB

    const int tid  = threadIdx.x;
    const int lane = tid & 31;
    const int wave = tid >> 5;
    const int hi   = lane >> 4;      // half-wave: 0 -> lanes 0-15, 1 -> 16-31
    const int l16  = lane & 15;

    // Each wave owns 16 consecutive rows (NHWC-flat). 4096 rows per image,
    // 16 | 4096, so a strip never crosses an image boundary.
    const int rowbase = blockIdx.x * 128 + wave * 16;
    const int b   = rowbase >> 12;
    const int hw0 = rowbase & 4095;

    // Resident A fragments: 16 rows x 256 dims of z in 64 x v2f = 128 VGPRs.
    // NCHW: z[b][d][hw] -> for fixed d, lanes 0-15 read 16 consecutive floats.
    const float* zb = z + (size_t)b * DDIM * HWSZ + hw0 + l16;
    v2f A[64];
    #pragma unroll
    for (int kk = 0; kk < 64; ++kk) {
        int d0 = kk * 4 + hi * 2;
        A[kk].x = zb[(size_t)d0 * HWSZ];
        A[kk].y = zb[(size_t)(d0 + 1) * HWSZ];
    }

    // Row squared norm: this lane covers dims {4k+2hi, 4k+2hi+1} of row l16;
    // the complementary half lives in lane (lane ^ 16).
    float z2 = 0.0f;
    #pragma unroll
    for (int kk = 0; kk < 64; ++kk)
        z2 += A[kk].x * A[kk].x + A[kk].y * A[kk].y;
    z2 += __shfl_xor(z2, 16, 32);     // full ||z_row||^2 for row l16

    float bestv[8];
    int   besti[8];
    #pragma unroll
    for (int j = 0; j < 8; ++j) { bestv[j] = 3.4e38f; besti[j] = 0; }

    for (int ch = 0; ch < KCODES / CHUNK; ++ch) {
        __syncthreads();   // previous chunk's compute done before overwrite
        // Cooperative stage: CHUNK codes x 256 dims, padded pitch LSTRIDE.
        const float* esrc = emb + (size_t)ch * CHUNK * DDIM;
        #pragma unroll
        for (int i = tid; i < CHUNK * DDIM; i += 256) {
            int c = i >> 8;          // code within chunk
            int k = i & 255;         // dim
            Bsh[c * LSTRIDE + k] = esrc[i];
        }
        __syncthreads();

        #pragma unroll
        for (int t = 0; t < CHUNK / 16; ++t) {
            const int cbase = ch * CHUNK + t * 16;
            // B fragment source for this lane: code = t*16+l16, k = kk*4+hi*2
            const float* bl = &Bsh[(t * 16 + l16) * LSTRIDE + hi * 2];

            v8f acc0 = {};
            v8f acc1 = {};
            #pragma unroll
            for (int kk = 0; kk < 64; kk += 2) {
                v2f b0 = *(const v2f*)(bl + kk * 4);
                v2f b1 = *(const v2f*)(bl + kk * 4 + 4);
                acc0 = __builtin_amdgcn_wmma_f32_16x16x4_f32(
                    false, A[kk],     false, b0, (short)0, acc0, false, false);
                acc1 = __builtin_amdgcn_wmma_f32_16x16x4_f32(
                    false, A[kk + 1], false, b1, (short)0, acc1, false, false);
            }

            const int   c  = cbase + l16;      // this lane's column (code id)
            const float hn = norms[c];         // 0.5*||e_c||^2
            #pragma unroll
            for (int j = 0; j < 8; ++j) {
                float score = hn - (acc0[j] + acc1[j]);   // argmin-equivalent
                if (score < bestv[j]) { bestv[j] = score; besti[j] = c; }
            }
        }
    }

    // Reduce over the 16 columns (lanes within each half-wave group).
    // C layout: lanes 0-15 hold rows j (VGPR j), lanes 16-31 hold rows 8+j.
    float lsum = 0.0f;    // this wave's loss partial (sum over its 16 rows)
    #pragma unroll
    for (int j = 0; j < 8; ++j) {
        float v  = bestv[j];
        int   id = besti[j];
        #pragma unroll
        for (int m = 8; m >= 1; m >>= 1) {
            float ov  = __shfl_xor(v,  m, 32);
            int   oid = __shfl_xor(id, m, 32);
            if (ov < v || (ov == v && oid < id)) { v = ov; id = oid; }
        }
        // Row for this group: r = hi*8 + j. The lane with l16 == r owns that
        // row's ||z||^2, so let it be the writer (v,id are group-uniform).
        const int r = hi * 8 + j;
        if (l16 == r) {
            int row = rowbase + r;
            idx_ws[row]  = id;
            idx_out[row] = (float)id;
            lsum += z2 + 2.0f * v;   // = ||z_q - z||^2 for this row
        }
    }

    // One atomic per wave for the loss.
    #pragma unroll
    for (int m = 16; m >= 1; m >>= 1) lsum += __shfl_xor(lsum, m, 32);
    if (lane == 0)
        atomicAdd(loss_out, lsum * (BETA_F / (float)NELEM));
}

// ---------------- Phase 2: write-only gather of z_q into NCHW out ----------
__global__ __launch_bounds__(256) void vq_gather_kernel(
    const float* __restrict__ emb, const int* __restrict__ idx_ws,
    float* __restrict__ out)
{
    int o  = blockIdx.x * 256 + threadIdx.x;   // flat NCHW index, < NELEM
    int hw = o & 4095;
    int t  = o >> 12;
    int d  = t & 255;
    int b  = t >> 8;
    int n  = (b << 12) + hw;                   // NHWC-flat row

    int id = idx_ws[n];                        // coalesced
    out[o] = emb[(size_t)id * DDIM + d];       // gather from L2-resident table
}

// ---------------------------------------------------------------------------
extern "C" void kernel_launch(void* const* d_in, const int* in_sizes, int n_in,
                              void* d_out, int out_size, void* d_ws, size_t ws_size,
                              hipStream_t stream)
{
    const float* z   = (const float*)d_in[0];   // [16,256,64,64] fp32
    const float* emb = (const float*)d_in[1];   // [1024,256]     fp32

    float* out      = (float*)d_out;            // 16,777,216 floats (NCHW)
    float* idx_out  = out + NELEM;              // 65,536 indices (as float)
    float* loss_out = idx_out + NROWS;          // 1 scalar

    float* norms  = (float*)d_ws;                       // 4 KB
    int*   idx_ws = (int*)((char*)d_ws + 4096);         // 256 KB

    vq_norms_kernel <<<KCODES / 256, 256, 0, stream>>>(emb, norms, loss_out);
    vq_argmin_kernel<<<NROWS / 128,  256, 0, stream>>>(z, emb, norms, idx_ws,
                                                       idx_out, loss_out);
    vq_gather_kernel<<<NELEM / 256,  256, 0, stream>>>(emb, idx_ws, out);
}